// SpatialSmoothnessLoss_32057635897736
// MI455X (gfx1250) — compile-verified
//
#include <hip/hip_runtime.h>

// SpatialSmoothnessLoss for MI455X (gfx1250, wave32).
// Distance tiles via V_WMMA_F32_16X16X4_F32 with |cand|^2 folded into the
// K-dim (A=(x,y,z,|c|^2), B=(-2x',-2y',-2z',1)). Radius-gated register
// top-16 per lane (value-equality replace, distinct sentinels), peeled
// self-tile, per-tile min-gate, shfl_xor half-merge, fused feature gather.

typedef float v2f __attribute__((ext_vector_type(2)));
typedef float v8f __attribute__((ext_vector_type(8)));

#define NK          16
#define RAD2        0.01f      // RADIUS^2
#define NPTS        12288
#define DFEAT       32
#define CHUNK       3072       // candidate points staged in LDS per pass
#define THREADS     256        // 8 waves
#define ROWS_PER_BLOCK 128     // 16 rows per wave * 8 waves
#define SENT(k)     (1.0e30f * (1.0f + 0.01f * (float)(k)))  // distinct, > RAD2
#define SELF_MARK   3.0e38f

// Replace the (unique) slot holding the current max, then refresh max/gate.
#define INSERT(D2, IDX)                                                   \
    do {                                                                  \
        _Pragma("unroll")                                                 \
        for (int k_ = 0; k_ < NK; ++k_) {                                 \
            const bool h_ = (bd[k_] == cm);                               \
            bd[k_] = h_ ? (D2) : bd[k_];                                  \
            bi[k_] = h_ ? (IDX) : bi[k_];                                 \
        }                                                                 \
        cm = bd[0];                                                       \
        _Pragma("unroll")                                                 \
        for (int k_ = 1; k_ < NK; ++k_) cm = fmaxf(cm, bd[k_]);           \
        cmEff = fminf(cm, RAD2);                                          \
    } while (0)

// One 16-candidate tile: 1 ds_load_b64 + 1 wmma + 8 adds + min-tree + gate.
#define PROC_TILE(M0, C0, SELFCHK)                                        \
    do {                                                                  \
        const float2 av_ = s_ab[pbase + (M0) + lhalf];                    \
        v2f amat_; amat_.x = av_.x; amat_.y = av_.y;                      \
        v8f c_ = {};                                                      \
        c_ = __builtin_amdgcn_wmma_f32_16x16x4_f32(                       \
                 false, amat_, false, bmat, (short)0, c_, false, false);  \
        float d2v_[8];                                                    \
        _Pragma("unroll")                                                 \
        for (int r_ = 0; r_ < 8; ++r_) {                                  \
            float d2_ = own_sq + c_[r_];                                  \
            if (SELFCHK) d2_ = (r_ == selfR) ? SELF_MARK : d2_;           \
            d2v_[r_] = d2_;                                               \
        }                                                                 \
        const float dmin_ =                                               \
            fminf(fminf(fminf(d2v_[0], d2v_[1]), fminf(d2v_[2], d2v_[3])),\
                  fminf(fminf(d2v_[4], d2v_[5]), fminf(d2v_[6], d2v_[7])));\
        if (dmin_ < cmEff) {                                              \
            const int candBase_ = (C0) + (M0) + candHalf;                 \
            _Pragma("unroll")                                             \
            for (int r_ = 0; r_ < 8; ++r_) {                              \
                if (d2v_[r_] < cmEff) INSERT(d2v_[r_], candBase_ + r_);   \
            }                                                             \
        }                                                                 \
    } while (0)

__global__ __launch_bounds__(THREADS)
void knn_smooth_kernel(const float* __restrict__ pred,
                       const float* __restrict__ coord,
                       const int*   __restrict__ vmask,
                       float* __restrict__ smooth_out)
{
    // Pair layout: s_ab[t]       = (x[t], y[t])      -> A K0,K1 (lanes 0-15)
    //              s_ab[CHUNK+t] = (z[t], |c[t]|^2)  -> A K2,K3 (lanes 16-31)
    __shared__ float2 s_ab[2 * CHUNK];

    const int   tid   = threadIdx.x;
    const int   lane  = tid & 31;
    const int   wave  = tid >> 5;
    const int   lhalf = lane & 15;
    const bool  lo    = lane < 16;
    const int   rowBase = blockIdx.x * ROWS_PER_BLOCK + wave * 16;
    const int   myRow   = rowBase + lhalf;          // this lane's row (N dim)
    const float INF     = __builtin_inff();

    // B operand (own rows, 4x16, K = {-2x',-2y',-2z',1}) — loop invariant.
    const float rx = coord[3*myRow+0];
    const float ry = coord[3*myRow+1];
    const float rz = coord[3*myRow+2];
    const float own_sq = rx*rx + ry*ry + rz*rz;
    v2f bmat;
    bmat.x = lo ? (-2.0f * rx) : (-2.0f * rz);
    bmat.y = lo ? (-2.0f * ry) : 1.0f;

    const int pbase    = lo ? 0 : CHUNK;   // float2 base for this lane's A half
    const int candHalf = lo ? 0 : 8;       // M offset of this lane's candidates
    const int selfR    = lhalf - candHalf; // self hit iff r == selfR (chunk-inv)

    // Self tile location (uniform per wave: rowBase is 16-aligned).
    const int selfChunk0 = (rowBase / CHUNK) * CHUNK;
    const int m0self     = rowBase % CHUNK;

    // Radius-gated top-16 (squared distances) over this lane's half-partition.
    float bd[NK];
    int   bi[NK];
#pragma unroll
    for (int k = 0; k < NK; ++k) { bd[k] = SENT(k); bi[k] = -1; }
    float cm    = SENT(NK - 1);  // max of kept entries (largest sentinel)
    float cmEff = RAD2;          // insertion gate = min(cm, R^2)

    for (int chunk0 = 0; chunk0 < NPTS; chunk0 += CHUNK) {
        __syncthreads();
        for (int t = tid; t < CHUNK; t += THREADS) {
            const int g = chunk0 + t;
            const float x = coord[3*g+0];
            const float y = coord[3*g+1];
            const float z = coord[3*g+2];
            const float sq = x*x + y*y + z*z;
            s_ab[t]         = make_float2(x, y);
            s_ab[CHUNK + t] = make_float2(z, (vmask[g] > 0) ? sq : INF);
        }
        __syncthreads();

        if (chunk0 == selfChunk0) {
#pragma unroll 2
            for (int m0 = 0; m0 < m0self; m0 += 16) PROC_TILE(m0, chunk0, 0);
            PROC_TILE(m0self, chunk0, 1);
#pragma unroll 2
            for (int m0 = m0self + 16; m0 < CHUNK; m0 += 16) PROC_TILE(m0, chunk0, 0);
        } else {
#pragma unroll 2
            for (int m0 = 0; m0 < CHUNK; m0 += 16) PROC_TILE(m0, chunk0, 0);
        }
    }

    // Merge with partner lane (lane ^ 16): the two half-partition top-16s
    // jointly cover every candidate that can contribute.
    float md[2*NK];
    int   mi[2*NK];
#pragma unroll
    for (int k = 0; k < NK; ++k) {
        md[k]      = bd[k];
        mi[k]      = bi[k];
        md[k + NK] = __shfl_xor(bd[k], 16, 32);
        mi[k + NK] = __shfl_xor(bi[k], 16, 32);
    }

    // Extract the 16 smallest of the merged 32 (both half-lanes see the same
    // multiset, so each lane ends with the full final list for its row).
    float fd[NK];
    int   fi[NK];
#pragma unroll
    for (int j = 0; j < NK; ++j) {
        float m = md[0];
#pragma unroll
        for (int k = 1; k < 2*NK; ++k) m = fminf(m, md[k]);
        bool done = false;
        fi[j] = -1;
#pragma unroll
        for (int k = 0; k < 2*NK; ++k) {
            const bool h = (!done) && (md[k] == m);
            fi[j] = h ? mi[k] : fi[j];
            md[k] = h ? INF   : md[k];
            done  = done || h;
        }
        fd[j] = m;
    }

    // Feature phase: lane handles neighbors 0..7 (lo) or 8..15 (hi).
    const float4* pv = (const float4*)pred;
    float4 p[DFEAT/4];
#pragma unroll
    for (int c4 = 0; c4 < DFEAT/4; ++c4) p[c4] = pv[myRow * (DFEAT/4) + c4];

    float acc = 0.0f, cnt = 0.0f;
#pragma unroll
    for (int k = 0; k < 8; ++k) {
        const float ndk = lo ? fd[k] : fd[k + 8];
        const int   nik = lo ? fi[k] : fi[k + 8];
        if (ndk < RAD2) {   // within radius (sentinel/padding entries fail)
            cnt += 1.0f;
#pragma unroll
            for (int c4 = 0; c4 < DFEAT/4; ++c4) {
                const float4 q = pv[nik * (DFEAT/4) + c4];
                const float dx = p[c4].x - q.x;
                const float dy = p[c4].y - q.y;
                const float dz = p[c4].z - q.z;
                const float dw = p[c4].w - q.w;
                acc += dx*dx + dy*dy + dz*dz + dw*dw;
            }
        }
    }
    acc += __shfl_xor(acc, 16, 32);
    cnt += __shfl_xor(cnt, 16, 32);

    if (lo) {
        float s = acc / fmaxf(cnt, 1.0f);
        s = (vmask[myRow] > 0) ? s : 0.0f;
        smooth_out[myRow] = s;
    }
}

__global__ __launch_bounds__(256)
void reduce_kernel(const float* __restrict__ s,
                   const int*   __restrict__ vmask,
                   float* __restrict__ out, int n)
{
    __shared__ float sh[256], shv[256];
    const int tid = threadIdx.x;
    float a = 0.0f, b = 0.0f;
    for (int i = tid; i < n; i += 256) {
        a += s[i];
        b += (vmask[i] > 0) ? 1.0f : 0.0f;
    }
    sh[tid] = a; shv[tid] = b;
    __syncthreads();
    for (int st = 128; st > 0; st >>= 1) {
        if (tid < st) { sh[tid] += sh[tid + st]; shv[tid] += shv[tid + st]; }
        __syncthreads();
    }
    if (tid == 0) out[0] = 1.0f * (sh[0] / fmaxf(shv[0], 1.0f));  // LOSS_WEIGHT=1
}

extern "C" void kernel_launch(void* const* d_in, const int* in_sizes, int n_in,
                              void* d_out, int out_size, void* d_ws, size_t ws_size,
                              hipStream_t stream) {
    (void)in_sizes; (void)n_in; (void)out_size; (void)ws_size;
    const float* pred  = (const float*)d_in[0];
    // d_in[1] = target (unused by the reference)
    const int*   vmask = (const int*)d_in[2];
    const float* coord = (const float*)d_in[3];

    float* smooth = (float*)d_ws;   // N floats of scratch

    knn_smooth_kernel<<<NPTS / ROWS_PER_BLOCK, THREADS, 0, stream>>>(
        pred, coord, vmask, smooth);
    reduce_kernel<<<1, 256, 0, stream>>>(smooth, vmask, (float*)d_out, NPTS);
}